// Network_13804024889408
// MI455X (gfx1250) — compile-verified
//
#include <hip/hip_runtime.h>
#include <hip/hip_bf16.h>

typedef __attribute__((ext_vector_type(2))) float v2f;
typedef __attribute__((ext_vector_type(8))) float v8f;

#define DCH 10  // channel count D
// feat channel order: [dis, relx, rely, relz, cx, cy, cz, nx, ny, nz], pad 10,11 = 0

__global__ __launch_bounds__(256) void attpool_wmma_kernel(
    const float* __restrict__ xyz,       // (B,N,3)
    const float* __restrict__ fc_w,      // (10,10)
    const float* __restrict__ mlp_w,     // (10,10)
    const float* __restrict__ bn_gamma,
    const float* __restrict__ bn_beta,
    const float* __restrict__ bn_mean,
    const float* __restrict__ bn_var,
    const int*   __restrict__ neigh_idx, // (B,N,16)
    float*       __restrict__ y_out,     // (B,N,10) scratch
    int B, int N)
{
  // 8 waves/block, one point per wave.
  __shared__ float lfeat[8 * 192 + 16]; // per-wave 16 rows x 12 floats
  __shared__ float sBt[192];            // sBt[o*12+i] = fc_w[o][i], zero-padded

  const int lane = threadIdx.x & 31;
  const int wid  = threadIdx.x >> 5;
  const int p    = blockIdx.x * 8 + wid;       // global point id, grid sized exactly
  const int b    = p / N;
  const int n    = p - b * N;
  const int m    = lane & 15;                  // neighbor row (A) / channel col (B,C)
  const int h    = lane >> 4;                  // half-wave select

  // stage padded fc_w (transposed-pair layout) once per block
  if (threadIdx.x < 192) {
    const int o = threadIdx.x / 12;            // out channel 0..15
    const int i = threadIdx.x % 12;            // inner dim 0..11
    sBt[threadIdx.x] = (i < DCH && o < DCH) ? fc_w[o * DCH + i] : 0.0f;
  }

  // center point (uniform across wave)
  const size_t pbase = (size_t)(b * (size_t)N + n);
  const float cx = xyz[pbase * 3 + 0];
  const float cy = xyz[pbase * 3 + 1];
  const float cz = xyz[pbase * 3 + 2];

  // gather neighbor m (lanes m and m+16 duplicate the load -> L0/L2 hit)
  const int nb = neigh_idx[pbase * 16 + m];
  const size_t nbase = (size_t)(b * (size_t)N + nb);
  const float nx = xyz[nbase * 3 + 0];
  const float ny = xyz[nbase * 3 + 1];
  const float nz = xyz[nbase * 3 + 2];

  const float rx = cx - nx, ry = cy - ny, rz = cz - nz;
  const float dis = sqrtf(rx * rx + ry * ry + rz * rz);

  // stage feat row m into LDS; lanes m and m+16 write identical values (benign)
  float* fl = &lfeat[wid * 192];
  float* arow = fl + m * 12;                   // 48B row stride, 16B aligned
  {
    float4* fl4 = (float4*)arow;
    fl4[0] = make_float4(dis, rx, ry, rz);
    fl4[1] = make_float4(cx, cy, cz, nx);
    fl4[2] = make_float4(ny, nz, 0.0f, 0.0f);
  }
  __syncthreads();   // covers both sBt and lfeat staging

  // att = feat(16x12) * fc_w^T(12x16): three chained V_WMMA_F32_16X16X4_F32
  // A 16x4 layout: lane holds A[m][k0], A[m][k0+1] with k0 = 4*c0 + 2*h
  // B 4x16 layout: lane holds B[k0][m], B[k0+1][m] = fc_w[m][k0], fc_w[m][k0+1]
  const float* brow = sBt + m * 12;
  v8f acc = {};
#pragma unroll
  for (int c0 = 0; c0 < 3; ++c0) {
    const int k0 = 4 * c0 + 2 * h;
    const v2f a  = *(const v2f*)(arow + k0);   // ds_load_b64 (8B aligned)
    const v2f bb = *(const v2f*)(brow + k0);   // ds_load_b64 (8B aligned)
    acc = __builtin_amdgcn_wmma_f32_16x16x4_f32(
        /*neg_a=*/false, a, /*neg_b=*/false, bb,
        /*c_mod=*/(short)0, acc, /*reuse_a=*/false, /*reuse_b=*/false);
  }
  // C layout: acc[r] = att[k = r + 8h][channel n = m]

  // softmax over k (16 values split across the two half-waves)
  float mx = acc[0];
#pragma unroll
  for (int r = 1; r < 8; ++r) mx = fmaxf(mx, acc[r]);
  mx = fmaxf(mx, __shfl_xor(mx, 16, 32));

  float e[8];
  float s = 0.0f;
#pragma unroll
  for (int r = 0; r < 8; ++r) { e[r] = __expf(acc[r] - mx); s += e[r]; }
  s += __shfl_xor(s, 16, 32);
  const float inv = 1.0f / s;

  // agg[n] = sum_k feat[k][n] * scores[k][n]  (read feat transposed from LDS)
  float part = 0.0f;
#pragma unroll
  for (int r = 0; r < 8; ++r) {
    const float fvk = fl[(r + 8 * h) * 12 + m];
    part = fmaf(fvk, e[r] * inv, part);
  }
  const float agg = part + __shfl_xor(part, 16, 32); // lanes n and n+16 hold agg[n]

  // y[o] = sum_i mlp_w[o][i] * agg[i]; broadcast agg[i] from lane i.
  // Lanes with m >= 10 compute a dead value (store below is guarded) -> clamp row.
  const float* mrow = mlp_w + (m < DCH ? m : DCH - 1) * DCH;
  float yo = 0.0f;
#pragma unroll
  for (int i = 0; i < DCH; ++i) {
    const float ai = __shfl(agg, i, 32);
    yo = fmaf(mrow[i], ai, yo);
  }

  if (h == 0 && m < DCH) {
    const float g  = bn_gamma[m];
    const float bt = bn_beta[m];
    const float mu = bn_mean[m];
    const float vv = bn_var[m];
    float r = (yo - mu) * rsqrtf(vv + 1e-5f) * g + bt;
    r = fmaxf(r, 0.0f);
    y_out[(size_t)p * DCH + m] = r;
  }
}

__global__ __launch_bounds__(256) void maxpool_gather_kernel(
    const float* __restrict__ y,        // (B,N,10)
    const int*   __restrict__ sub_idx,  // (B,M,16)
    float*       __restrict__ out,      // (B,M,10)
    int B, int N, int M)
{
  const int t = blockIdx.x * 256 + threadIdx.x;
  const int total = B * M * DCH;
  if (t >= total) return;
  const int o  = t % DCH;
  const int bm = t / DCH;                 // b*M + m
  const int b  = bm / M;
  const int* si = sub_idx + (size_t)bm * 16;
  const size_t ybase = (size_t)b * (size_t)N * DCH;
  float mx = -3.402823466e38f;
#pragma unroll
  for (int k = 0; k < 16; ++k) {
    const int idx = si[k];
    mx = fmaxf(mx, y[ybase + (size_t)idx * DCH + o]);
  }
  out[t] = mx;
}

extern "C" void kernel_launch(void* const* d_in, const int* in_sizes, int n_in,
                              void* d_out, int out_size, void* d_ws, size_t ws_size,
                              hipStream_t stream) {
  const float* xyz       = (const float*)d_in[0];
  const float* fc_w      = (const float*)d_in[1];
  const float* mlp_w     = (const float*)d_in[2];
  const float* bn_gamma  = (const float*)d_in[3];
  const float* bn_beta   = (const float*)d_in[4];
  const float* bn_mean   = (const float*)d_in[5];
  const float* bn_var    = (const float*)d_in[6];
  const int*   neigh_idx = (const int*)d_in[7];
  const int*   sub_idx   = (const int*)d_in[8];
  float*       out       = (float*)d_out;
  float*       y_ws      = (float*)d_ws;      // B*N*10 floats = 5.24 MB

  const int B = 2;
  const int N = in_sizes[7] / (B * 16);       // 65536
  const int M = in_sizes[8] / (B * 16);       // 16384

  const int points = B * N;                   // 131072, 8 waves/block
  attpool_wmma_kernel<<<points / 8, 256, 0, stream>>>(
      xyz, fc_w, mlp_w, bn_gamma, bn_beta, bn_mean, bn_var, neigh_idx,
      y_ws, B, N);

  const int total = B * M * DCH;
  maxpool_gather_kernel<<<(total + 255) / 256, 256, 0, stream>>>(
      y_ws, sub_idx, out, B, N, M);
}